// DtwLayer_20976620273718
// MI455X (gfx1250) — compile-verified
//
#include <hip/hip_runtime.h>
#include <hip/hip_bf16.h>

#define GAMMA_F   0.2f
#define BIGV      1e8f
#define NN        256
#define MM        256
#define OO        64
#define BAND      128                   // bandwidth * max(N,M) = 0.5*256
#define K_EXP2    7.213475204444817f    // (1/gamma) * log2(e)
#define G_LN2     0.13862943611198906f  // gamma * ln(2)

typedef __attribute__((ext_vector_type(2))) float v2f;
typedef __attribute__((ext_vector_type(8))) float v8f;

// One block per (b,o) pair: 256 threads = one thread per DP row.
// Anti-diagonal wavefront sweep with one barrier per diagonal.
// cost[i,j] = w_j*(x_i - y_j)^2 is a rank-3 GEMM computed in 16x16 tiles by
// V_WMMA_F32_16X16X4_F32 into a 2-deep tile-diagonal LDS ping-pong; the matrix
// pipe (XDL, co-executes like TRANS ops) overlaps with the exp/log softmin.
__launch_bounds__(256)
__global__ void sdtw_wavefront_kernel(const float* __restrict__ x,      // [B,N]
                                      const float* __restrict__ align,  // [O,M]
                                      const float* __restrict__ bias,   // [O,1]
                                      const float* __restrict__ weight, // [O,M]
                                      float* __restrict__ out)          // [B,O]
{
    __shared__ float  costBuf[2][16][16][16]; // 32 KB: two tile-anti-diagonals of cost
    __shared__ float2 diagBuf[2][NN];         // 4 KB : (R[d-1][i], R[d-2][i]) ping-pong
    __shared__ float  xs[NN];                 // x row for this b
    __shared__ float  wArr[NN];               // w_j = exp(weight)
    __shared__ float  m2wy[NN];               // -2*w_j*y_j
    __shared__ float  wy2[NN];                // w_j*y_j^2

    const int tid = threadIdx.x;
    const int bo  = blockIdx.x;          // bo = b*O + o  (row-major [B,O] output)
    const int b   = bo >> 6;
    const int o   = bo & (OO - 1);

    // ---- preload: x row + rank-3 factors of the cost matrix ----
    {
        xs[tid] = x[b * NN + tid];
        float yv = align[o * MM + tid] + bias[o];
        float ww = __expf(weight[o * MM + tid]);
        wArr[tid] = ww;
        m2wy[tid] = -2.0f * ww * yv;
        wy2[tid]  = ww * yv * yv;
    }
    __syncthreads();

    const int  wave = tid >> 5;
    const int  lane = tid & 31;
    const int  lr   = lane & 15;
    const bool hi   = lane >= 16;

    float r1 = BIGV;   // R[d-1][tid]
    float r2 = BIGV;   // R[d-2][tid]

    for (int g = 0; g < 32; ++g) {
        // ---- WMMA phase: cost tiles with ti+tj == g into costBuf[g&1] ----
        if (g <= 30) {
#pragma unroll
            for (int rep = 0; rep < 2; ++rep) {
                int ti = wave + rep * 8;         // 8 waves cover up to 16 tiles
                int tj = g - ti;
                if (tj >= 0 && tj < 16) {        // wave-uniform -> EXEC all-1s at WMMA
                    // A (16x4): row m = 16*ti+lr; K0=x^2, K1=x, K2=1, K3=0
                    float xv = xs[ti * 16 + lr];
                    v2f a;
                    a.x = hi ? 1.0f : xv * xv;   // VGPR0: K0 (lanes 0-15) / K2 (16-31)
                    a.y = hi ? 0.0f : xv;        // VGPR1: K1 / K3
                    // B (4x16): col n = 16*tj+lr; K0=w, K1=-2wy, K2=wy^2, K3=0
                    int jj = tj * 16 + lr;
                    v2f bf;
                    bf.x = hi ? wy2[jj] : wArr[jj];
                    bf.y = hi ? 0.0f    : m2wy[jj];
                    v8f acc = {};
                    acc = __builtin_amdgcn_wmma_f32_16x16x4_f32(
                        false, a, false, bf, (short)0, acc, false, false);
                    // C layout: vgpr r -> M = r (lanes 0-15) or 8+r (lanes 16-31), N = lr
                    int mbase = hi ? 8 : 0;
#pragma unroll
                    for (int r = 0; r < 8; ++r)
                        costBuf[g & 1][ti][mbase + r][lr] = acc[r];
                }
            }
        }

        // ---- wavefront sweep: diagonals d = 16g .. min(16g+15, 510) ----
        const int dend = (g == 31) ? 15 : 16;
        for (int dd = 0; dd < dend; ++dd) {
            const int d = g * 16 + dd;
            diagBuf[d & 1][tid] = make_float2(r1, r2);
            __syncthreads();   // orders diag exchange AND cost-tile stores vs loads

            // branchless neighbor fetch (tid==0 reads its own slot, then selects)
            const int nidx = (tid == 0) ? 0 : tid - 1;
            float2 nb = diagBuf[d & 1][nidx];
            float up = (tid == 0) ? BIGV : nb.x;                         // R[i-1, j]
            float dg = (tid == 0) ? ((d == 0) ? 0.0f : BIGV) : nb.y;     // R[i-1, j-1]
            const float lf = r1;                                         // R[i,   j-1]

            const int j = d - tid;
            float rnew = BIGV;
            if (j >= 0 && j < MM && abs(tid - j) <= BAND) {
                const int tdg = ((tid >> 4) + (j >> 4)) & 1;  // tile-diag parity
                float c = costBuf[tdg][tid >> 4][tid & 15][j & 15];
                // softmin_g(a,b,c) = m - (g*ln2)*log2( sum 2^((m - .)*K) ), m = min
                // v_exp_f32 is natively 2^x; v_log_f32 is natively log2(x)
                float m = fminf(dg, fminf(up, lf));
                float s = __builtin_amdgcn_exp2f((m - dg) * K_EXP2)
                        + __builtin_amdgcn_exp2f((m - up) * K_EXP2)
                        + __builtin_amdgcn_exp2f((m - lf) * K_EXP2);
                rnew = c + m - G_LN2 * __builtin_amdgcn_logf(s);
            }
            r2 = r1;
            r1 = rnew;
        }
    }

    if (tid == NN - 1)
        out[bo] = r1;      // R[N-1, M-1]
}

extern "C" void kernel_launch(void* const* d_in, const int* in_sizes, int n_in,
                              void* d_out, int out_size, void* d_ws, size_t ws_size,
                              hipStream_t stream) {
    (void)in_sizes; (void)n_in; (void)d_ws; (void)ws_size; (void)out_size;
    const float* x      = (const float*)d_in[0];  // input      [16,256]
    const float* align  = (const float*)d_in[1];  // align_seq  [64,256]
    const float* bias   = (const float*)d_in[2];  // bias       [64,1]
    const float* weight = (const float*)d_in[3];  // weight     [64,256]
    float* out = (float*)d_out;                   // [16,64]

    sdtw_wavefront_kernel<<<dim3(16 * OO), dim3(256), 0, stream>>>(
        x, align, bias, weight, out);
}